// KernelNorm2d_72164040507639
// MI455X (gfx1250) — compile-verified
//
#include <hip/hip_runtime.h>
#include <stdint.h>

// KernelNorm2d, kernel=stride=2x2, pad=0, eps=1e-5
// x[16,64,256,256] f32 -> same-shape f32; per (b, 2x2 tile) normalize over C*2*2=256 elems.
//
// Memory-bound (0.54 GB @ 23.3 TB/s ~= 23us). Strategy: one WG per (batch,row-pair),
// stage the 128KB tile global->LDS with CDNA5 async-to-LDS DMA, reduce in LDS,
// normalize from LDS, stream out with non-temporal b128 stores.

typedef float v4f __attribute__((ext_vector_type(4)));
typedef float v2f __attribute__((ext_vector_type(2)));

#define CC 64
#define HH 256
#define WW 256
// per-block tile: 64 ch x 2 rows x 256 w = 32768 floats = 128 KB

__global__ __launch_bounds__(256) void KernelNorm2d_gfx1250_kernel(
    const float* __restrict__ x, float* __restrict__ out) {
  __shared__ float sdata[CC * 2 * WW];   // 128 KB staged tile, layout: [cr=2c+r][w]
  __shared__ float p1[256], p2[256];     // per-thread partial sums
  __shared__ float s_mu[128], s_rs[128]; // per-window mean / rstd

  const int t   = threadIdx.x;
  const int blk = blockIdx.x;
  const int b   = blk >> 7;    // batch
  const int i   = blk & 127;   // tile-row (rows 2i, 2i+1)

  const size_t base_elem = (size_t)b * (CC * HH * WW) + (size_t)(2 * i) * WW;
  const uint64_t base_u64 = (uint64_t)(uintptr_t)(x + base_elem);

  // ---- Phase 1: async DMA global -> LDS (32 x b128 per thread = 128 KB/WG) ----
  // element index e = k*1024 + t*4 ; chunk = e/512 (one channel, rows 2i & 2i+1
  // are contiguous: 512 floats), inchunk = e%512. Global offset bytes:
  //   chunk*H*W*4 + inchunk*4 ; LDS offset bytes: e*4 (linear copy).
  {
    uint32_t ldsa = (uint32_t)(uintptr_t)(&sdata[0]) + (uint32_t)t * 16u;
    uint32_t voff = (uint32_t)(t >> 7) * (HH * WW * 4u) + (uint32_t)(t & 127) * 16u;
#pragma unroll
    for (int k = 0; k < 32; ++k) {
      asm volatile("global_load_async_to_lds_b128 %0, %1, %2"
                   :: "v"(ldsa), "v"(voff), "s"(base_u64)
                   : "memory");
      ldsa += 4096u;                 // +1024 floats in LDS
      voff += 2u * HH * WW * 4u;     // +2 channels in global
    }
  }
  asm volatile("s_wait_asynccnt 0x0" ::: "memory");
  __syncthreads();

  // ---- Phase 2: per-window sums. Window j covers columns w=2j,2j+1 over cr=0..127.
  // Thread t: j = t&127, half = t>>7 handles cr in [64*half, 64*half+64).
  {
    const int j    = t & 127;
    const int half = t >> 7;
    const float* colp = &sdata[(size_t)half * 64 * 256 + 2 * j];
    float s1 = 0.f, s2 = 0.f;
#pragma unroll 8
    for (int cr = 0; cr < 64; ++cr) {
      v2f v = *(const v2f*)(colp + (size_t)cr * 256);   // conflict-free ds_load_b64
      s1 += v.x + v.y;
      s2 += v.x * v.x + v.y * v.y;
    }
    p1[t] = s1;
    p2[t] = s2;
  }
  __syncthreads();

  if (t < 128) {
    const float inv_n = 1.0f / 256.0f;
    float S1 = p1[t] + p1[t + 128];
    float S2 = p2[t] + p2[t + 128];
    float mean = S1 * inv_n;
    float var  = S2 * inv_n - mean * mean;
    s_mu[t] = mean;
    s_rs[t] = 1.0f / sqrtf(var + 1e-5f);
  }
  __syncthreads();

  // ---- Phase 3: normalize from LDS, stream out (NT b128 stores). ----
  // Thread t, iter k reads LDS floats [k*1024 + t*4 .. +3]; those have fixed
  // w = (t&63)*4..+3 -> windows j0=(t&63)*2 (x,y) and j0+1 (z,w) for every k.
  {
    const int j0 = (t & 63) * 2;
    const float mu0 = s_mu[j0],     r0 = s_rs[j0];
    const float mu1 = s_mu[j0 + 1], r1 = s_rs[j0 + 1];
    float* op = out + base_elem + (size_t)(t >> 7) * (HH * WW) + (size_t)(t & 127) * 4;
    const float* ip = &sdata[(size_t)t * 4];
#pragma unroll
    for (int k = 0; k < 32; ++k) {
      v4f v = *(const v4f*)(ip + (size_t)k * 1024);
      v4f o;
      o.x = (v.x - mu0) * r0;
      o.y = (v.y - mu0) * r0;
      o.z = (v.z - mu1) * r1;
      o.w = (v.w - mu1) * r1;
      __builtin_nontemporal_store(o, (v4f*)(op + (size_t)k * 2 * HH * WW));
    }
  }
}

extern "C" void kernel_launch(void* const* d_in, const int* in_sizes, int n_in,
                              void* d_out, int out_size, void* d_ws, size_t ws_size,
                              hipStream_t stream) {
  const float* x = (const float*)d_in[0];
  float* out = (float*)d_out;
  (void)in_sizes; (void)n_in; (void)out_size; (void)d_ws; (void)ws_size;
  const int B = 16;
  const int nBlocks = B * (HH / 2);  // 16 * 128 = 2048 workgroups
  KernelNorm2d_gfx1250_kernel<<<dim3(nBlocks), dim3(256), 0, stream>>>(x, out);
}